// SubGraphs_56075093016968
// MI455X (gfx1250) — compile-verified
//
#include <hip/hip_runtime.h>
#include <hip/hip_bf16.h>
#include <math.h>

typedef __attribute__((ext_vector_type(2))) float v2f;
typedef __attribute__((ext_vector_type(8))) float v8f;
typedef __attribute__((ext_vector_type(4))) unsigned int v4u;
typedef __attribute__((ext_vector_type(8))) int v8i;
typedef __attribute__((ext_vector_type(4))) int v4i;

#define N_NODES 2048
#define K_SUB   16
#define F_IN    256
#define O_OUT   256

// ---------------------------------------------------------------------------
// TDM: async 2D tile load (32 rows x 64 f32, tensor row stride 2048 f32) into
// LDS with 2-DWORD padding every 64 DWORDs -> LDS row stride 66 floats
// (bank-conflict-free for the A-fragment read pattern).
// Descriptor layout per CDNA5 ISA ch.8 (D# groups 0/1; groups 2/3 = 0 for 2D).
// This toolchain exposes the 6-arg builtin (clang-23 / therock-10.0 headers).
// ---------------------------------------------------------------------------
__device__ __forceinline__ void tdm_load_adj_tile(const float* gsrc, unsigned lds_addr) {
    const unsigned long long ga = (unsigned long long)(uintptr_t)gsrc;
    v4u g0;
    g0.x = 1u;                                        // count=1, user mode
    g0.y = lds_addr;                                  // LDS byte address
    g0.z = (unsigned)(ga & 0xFFFFFFFFull);            // global_addr[31:0]
    g0.w = (unsigned)((ga >> 32) & 0x01FFFFFFull)     // global_addr[56:32]
         | 0x80000000u;                               // type = 2 ("image")
    v8i g1;
    g1[0] = (int)((2u << 16)                          // data_size = 4B
                | (1u << 20)                          // pad_enable
                | (5u << 22)                          // pad_interval: 64 DWORDs
                | (1u << 25));                        // pad_amount: 2 DWORDs
    g1[1] = (int)(2048u << 16);                       // tensor_dim0[15:0] (=2048)
    g1[2] = (int)(2048u << 16);                       // tensor_dim0 hi=0 | tensor_dim1 lo (=2048)
    g1[3] = (int)(64u << 16);                         // tensor_dim1 hi=0 | tile_dim0 = 64
    g1[4] = 32;                                       // tile_dim1 = 32, tile_dim2 = 0
    g1[5] = 2048;                                     // tensor_dim0_stride lo
    g1[6] = 0;                                        // stride0 hi | stride1 lo
    g1[7] = 0;
    const v4i zero4 = {};
    const v8i zero8 = {};
    __builtin_amdgcn_tensor_load_to_lds(g0, g1, zero4, zero4, zero8, 0);
}

// ---------------------------------------------------------------------------
// Kernel 1: partition logits + argmax -> gid[i]
// ---------------------------------------------------------------------------
__global__ __launch_bounds__(256)
void k_part(const float* __restrict__ x, const float* __restrict__ w,
            const float* __restrict__ b, int* __restrict__ gid) {
    const int i = blockIdx.x * 256 + threadIdx.x;
    float acc[K_SUB];
#pragma unroll
    for (int k = 0; k < K_SUB; ++k) acc[k] = b[k];
    for (int f = 0; f < F_IN; ++f) {
        const float xv = x[i * F_IN + f];
#pragma unroll
        for (int k = 0; k < K_SUB; ++k) acc[k] += xv * w[f * K_SUB + k];
    }
    int best = 0;
    float bv = fmaxf(acc[0], 0.0f);
#pragma unroll
    for (int k = 1; k < K_SUB; ++k) {
        const float t = fmaxf(acc[k], 0.0f);
        if (t > bv) { bv = t; best = k; }
    }
    gid[i] = best;
}

// ---------------------------------------------------------------------------
// Kernel 2: deterministic per-group compaction: list[k][..], cnt[k]
// ---------------------------------------------------------------------------
__global__ void k_lists(const int* __restrict__ gid, int* __restrict__ cnt,
                        int* __restrict__ list) {
    const int k = threadIdx.x;
    if (k >= K_SUB) return;
    int c = 0;
    for (int i = 0; i < N_NODES; ++i) {
        if (gid[i] == k) { list[k * N_NODES + c] = i; ++c; }
    }
    cnt[k] = c;
}

// ---------------------------------------------------------------------------
// Kernel 3a: per-row bucket sums of adj: rs[i][b] = sum_{j: gid[j]==b} adj[i][j]
// ---------------------------------------------------------------------------
__global__ __launch_bounds__(256)
void k_rowsum(const float* __restrict__ adj, const int* __restrict__ gid,
              float* __restrict__ rs) {
    __shared__ float part[8][32][17];
    const int tid = threadIdx.x, wave = tid >> 5, lane = tid & 31;
    const int i = blockIdx.x * 8 + wave;
    float acc[K_SUB];
#pragma unroll
    for (int b = 0; b < K_SUB; ++b) acc[b] = 0.0f;
    for (int it = 0; it < N_NODES / 32; ++it) {
        const int j = lane + it * 32;
        const float v = adj[i * N_NODES + j];
        const int g = gid[j];
#pragma unroll
        for (int b = 0; b < K_SUB; ++b) acc[b] += (g == b) ? v : 0.0f;
    }
#pragma unroll
    for (int b = 0; b < K_SUB; ++b) part[wave][lane][b] = acc[b];
    __syncthreads();
    if (tid < 128) {
        const int r = tid >> 4, b = tid & 15;
        float s = 0.0f;
#pragma unroll
        for (int l = 0; l < 32; ++l) s += part[r][l][b];
        rs[(blockIdx.x * 8 + r) * K_SUB + b] = s;
    }
}

// ---------------------------------------------------------------------------
// Kernel 3b: coarsen[a][b] = sum_{i: gid[i]==a} rs[i][b]   (deterministic)
// ---------------------------------------------------------------------------
__global__ void k_coarsen(const float* __restrict__ rs, const int* __restrict__ gid,
                          float* __restrict__ coarsen) {
    const int t = threadIdx.x;
    const int a = t >> 4, b = t & 15;
    float s = 0.0f;
    for (int i = 0; i < N_NODES; ++i)
        if (gid[i] == a) s += rs[i * K_SUB + b];
    coarsen[a * K_SUB + b] = s;
}

// ---------------------------------------------------------------------------
// Kernel 4: agg[N][F] = (adj .* same-group) @ x + 2*x     -- WMMA f32 16x16x4
// TDM double-buffered adj tiles (32x64) -> LDS (row stride 66 via D# padding),
// group mask applied at A-fragment read time.
// ---------------------------------------------------------------------------
__global__ __launch_bounds__(256)
void k_agg(const float* __restrict__ adj, const float* __restrict__ x,
           const int* __restrict__ gid, float* __restrict__ agg) {
    __shared__ float sbuf[2][32 * 66];
    const int tid = threadIdx.x;
    const int wave = tid >> 5, lane = tid & 31;
    const int i0 = blockIdx.x * 32;
    const int mt = wave & 1;
    const int ntb = (wave >> 1) * 4;
    const int m = lane & 15;
    const int kq = (lane >> 4) * 2;
    const int gi = gid[i0 + mt * 16 + m];       // this lane's output-row group

    v8f acc[4];
#pragma unroll
    for (int t = 0; t < 4; ++t) acc[t] = (v8f){};

    // prologue: kick off chunk 0
    if (tid < 32) {
        tdm_load_adj_tile(adj + (size_t)i0 * N_NODES,
                          (unsigned)(uintptr_t)&sbuf[0][0]);
    }

    for (int c = 0; c < N_NODES / 64; ++c) {
        const int p = c & 1;
        if (tid < 32) __builtin_amdgcn_s_wait_tensorcnt(0);
        __syncthreads();                         // tile p visible; buffer 1-p free
        if (tid < 32 && (c + 1) < N_NODES / 64) {
            tdm_load_adj_tile(adj + (size_t)i0 * N_NODES + (size_t)(c + 1) * 64,
                              (unsigned)(uintptr_t)&sbuf[c & 1 ? 0 : 1][0]);
        }
        const float* sa = &sbuf[p][0];
        const int j0 = c * 64;
        for (int kk = 0; kk < 64; kk += 4) {
            const int jr = j0 + kk + kq;
            v2f a;
            a.x = sa[(mt * 16 + m) * 66 + kk + kq];
            a.y = sa[(mt * 16 + m) * 66 + kk + kq + 1];
            a.x = (gi == gid[jr])     ? a.x : 0.0f;   // same-group mask
            a.y = (gi == gid[jr + 1]) ? a.y : 0.0f;
#pragma unroll
            for (int t = 0; t < 4; ++t) {
                const int f = (ntb + t) * 16 + m;
                v2f b;
                b.x = x[jr * F_IN + f];
                b.y = x[(jr + 1) * F_IN + f];
                acc[t] = __builtin_amdgcn_wmma_f32_16x16x4_f32(
                    false, a, false, b, (short)0, acc[t], false, false);
            }
        }
    }
    // epilogue: agg = acc + 2*x   (C/D layout: VGPR v -> M = v + 8*(lane>=16))
#pragma unroll
    for (int t = 0; t < 4; ++t) {
#pragma unroll
        for (int v = 0; v < 8; ++v) {
            const int row = i0 + mt * 16 + v + ((lane >> 4) ? 8 : 0);
            const int col = (ntb + t) * 16 + m;
            agg[row * F_IN + col] = acc[t][v] + 2.0f * x[row * F_IN + col];
        }
    }
}

// ---------------------------------------------------------------------------
// Kernel 5: per-group GCN: z = agg_rows @ W_k ; L2-normalize rows ; relu -> h
// ---------------------------------------------------------------------------
__global__ __launch_bounds__(256)
void k_gcn(const float* __restrict__ agg, const float* __restrict__ W,
           const int* __restrict__ cnt, const int* __restrict__ list,
           float* __restrict__ h) {
    const int k = blockIdx.y;
    const int t = blockIdx.x;
    const int cntk = cnt[k];
    if (t * 16 >= cntk) return;                 // uniform exit (before barriers)

    __shared__ float sA[16][258];
    __shared__ float zs[16][258];
    __shared__ int   nid[16];
    __shared__ float snorm[16];
    const int tid = threadIdx.x;

    if (tid < 16) {
        const int idx = t * 16 + tid;
        nid[tid] = (idx < cntk) ? list[k * N_NODES + idx] : -1;
    }
    __syncthreads();
#pragma unroll
    for (int e = 0; e < 16; ++e) {
        const int idx = tid + e * 256;
        const int r = idx >> 8, c = idx & 255;
        const int node = nid[r];
        sA[r][c] = (node >= 0) ? agg[node * F_IN + c] : 0.0f;
    }
    __syncthreads();

    const int wave = tid >> 5, lane = tid & 31;
    const int m = lane & 15, kq = (lane >> 4) * 2;
    const float* Wk = W + (size_t)k * F_IN * O_OUT;
    v8f acc0 = (v8f){}, acc1 = (v8f){};
    const int o0 = (wave * 2 + 0) * 16 + m;
    const int o1 = (wave * 2 + 1) * 16 + m;

    for (int kk = 0; kk < F_IN; kk += 4) {
        v2f a;
        a.x = sA[m][kk + kq];
        a.y = sA[m][kk + kq + 1];
        v2f b0, b1;
        b0.x = Wk[(kk + kq) * O_OUT + o0];
        b0.y = Wk[(kk + kq + 1) * O_OUT + o0];
        acc0 = __builtin_amdgcn_wmma_f32_16x16x4_f32(
            false, a, false, b0, (short)0, acc0, false, false);
        b1.x = Wk[(kk + kq) * O_OUT + o1];
        b1.y = Wk[(kk + kq + 1) * O_OUT + o1];
        acc1 = __builtin_amdgcn_wmma_f32_16x16x4_f32(
            false, a, false, b1, (short)0, acc1, false, false);
    }
#pragma unroll
    for (int v = 0; v < 8; ++v) {
        const int mr = v + ((lane >> 4) ? 8 : 0);
        zs[mr][(wave * 2 + 0) * 16 + m] = acc0[v];
        zs[mr][(wave * 2 + 1) * 16 + m] = acc1[v];
    }
    __syncthreads();
    if (tid < 16) {
        float ss = 0.0f;
        for (int o = 0; o < O_OUT; ++o) { const float z = zs[tid][o]; ss += z * z; }
        snorm[tid] = fmaxf(sqrtf(ss), 1e-12f);
    }
    __syncthreads();
#pragma unroll
    for (int e = 0; e < 16; ++e) {
        const int idx = tid + e * 256;
        const int r = idx >> 8, c = idx & 255;
        const int node = nid[r];
        if (node >= 0)
            h[node * O_OUT + c] = fmaxf(zs[r][c] / snorm[r], 0.0f);
    }
}

// ---------------------------------------------------------------------------
// Kernel 6: batch-norm column stats (deterministic single block)
// ---------------------------------------------------------------------------
__global__ void k_bnstats(const float* __restrict__ h, float* __restrict__ meanv,
                          float* __restrict__ invstd) {
    const int c = threadIdx.x;
    float s = 0.0f, s2 = 0.0f;
    for (int i = 0; i < N_NODES; ++i) {
        const float v = h[i * O_OUT + c];
        s += v; s2 += v * v;
    }
    const float mean = s * (1.0f / N_NODES);
    const float var  = s2 * (1.0f / N_NODES) - mean * mean;
    meanv[c] = mean;
    invstd[c] = rsqrtf(var + 1e-5f);
}

// ---------------------------------------------------------------------------
// Kernel 7: out[k][o] = sum_{i in group k} (h[i][o]-mean[o])*invstd[o]
// ---------------------------------------------------------------------------
__global__ void k_out(const float* __restrict__ h, const float* __restrict__ meanv,
                      const float* __restrict__ invstd, const int* __restrict__ cnt,
                      const int* __restrict__ list, float* __restrict__ out) {
    const int k = blockIdx.x;
    const int o = threadIdx.x;
    const float mean = meanv[o], is = invstd[o];
    const int c = cnt[k];
    float s = 0.0f;
    for (int idx = 0; idx < c; ++idx) {
        const int i = list[k * N_NODES + idx];
        s += (h[i * O_OUT + o] - mean) * is;
    }
    out[k * O_OUT + o] = s;
}

// ---------------------------------------------------------------------------
extern "C" void kernel_launch(void* const* d_in, const int* in_sizes, int n_in,
                              void* d_out, int out_size, void* d_ws, size_t ws_size,
                              hipStream_t stream) {
    (void)in_sizes; (void)n_in; (void)out_size; (void)ws_size;
    const float* x      = (const float*)d_in[0];   // [2048,256]
    const float* adj    = (const float*)d_in[1];   // [2048,2048]
    const float* w_part = (const float*)d_in[2];   // [256,16]
    const float* b_part = (const float*)d_in[3];   // [16]
    const float* gcn_w  = (const float*)d_in[4];   // [16,256,256]

    float* outp    = (float*)d_out;                // [16*256] out
    float* coarsen = outp + K_SUB * O_OUT;         // [16*16] coarsen_adj

    char* ws = (char*)d_ws;
    int*   gid    = (int*)(ws + 0);                // 8192 B
    int*   cnt    = (int*)(ws + 8192);             // 64 B
    int*   list   = (int*)(ws + 16384);            // 131072 B
    float* meanv  = (float*)(ws + 147456);         // 1024 B
    float* invstd = (float*)(ws + 148480);         // 1024 B
    float* rs     = (float*)(ws + 163840);         // 131072 B
    float* agg    = (float*)(ws + 327680);         // 2 MB
    float* h      = (float*)(ws + 2424832);        // 2 MB   (total ~4.3 MB)

    k_part   <<<N_NODES / 256, 256, 0, stream>>>(x, w_part, b_part, gid);
    k_lists  <<<1, 32, 0, stream>>>(gid, cnt, list);
    k_rowsum <<<N_NODES / 8, 256, 0, stream>>>(adj, gid, rs);
    k_coarsen<<<1, 256, 0, stream>>>(rs, gid, coarsen);
    k_agg    <<<N_NODES / 32, 256, 0, stream>>>(adj, x, gid, agg);
    k_gcn    <<<dim3(N_NODES / 16, K_SUB), 256, 0, stream>>>(agg, gcn_w, cnt, list, h);
    k_bnstats<<<1, 256, 0, stream>>>(h, meanv, invstd);
    k_out    <<<K_SUB, 256, 0, stream>>>(h, meanv, invstd, cnt, list, outp);
}